// GroupEncoder_34110630265158
// MI455X (gfx1250) — compile-verified
//
#include <hip/hip_runtime.h>
#include <hip/hip_bf16.h>

// Problem constants
#define NB    32      // batch
#define KTOT  16384   // I*J
#define XD    128     // X_DIM
#define RD    32      // R_DIM
#define UD    16
#define VD    8
#define HD    16

#define KSPLIT 16            // K chunks (grid.x) -> 1024 K per block
#define KPB   (KTOT/KSPLIT)  // K per block
#define KC    64             // K per LDS staging iteration

// Padded LDS row pitches (floats). Stride % 32 == 8, so rows k and k+2 are
// 16 banks apart -> lanes 0-15 (row k) and 16-31 (row k+2) use disjoint banks.
#define XP    136            // x tile row pitch (544 B, 16B-aligned)
#define RP    40             // r tile row pitch (160 B, 16B-aligned)

typedef __attribute__((ext_vector_type(2))) float v2f;
typedef __attribute__((ext_vector_type(8))) float v8f;

__global__ void ge_zero(float* p, int nelem) {
    int i = blockIdx.x * blockDim.x + threadIdx.x;
    if (i < nelem) p[i] = 0.0f;
}

// Direct global -> LDS async DMA (bypasses VGPRs), tracked by ASYNCcnt.
__device__ __forceinline__ void async_b128(unsigned lds_addr, const void* gaddr) {
    asm volatile("global_load_async_to_lds_b128 %0, %1, off"
                 :: "v"(lds_addr), "v"(gaddr) : "memory");
}

// y[n, m(=R_DIM row), x] += sum over this block's K slice of r[n,k,m]*x[n,k,x]
__global__ __launch_bounds__(512) void ge_stage1(const float* __restrict__ x,
                                                 const float* __restrict__ r,
                                                 float* __restrict__ y) {
    __shared__ __align__(16) float xs[2][KC][XP];  // 2 x 34 KB
    __shared__ __align__(16) float rs[2][KC][RP];  // 2 x 10 KB

    const int n      = blockIdx.y;
    const long kbase = (long)blockIdx.x * KPB;
    const float* xn  = x + (long)n * KTOT * XD;
    const float* rn  = r + (long)n * KTOT * RD;

    const int tid   = threadIdx.x;
    const int wave  = tid >> 5;
    const int lane  = tid & 31;
    const int half  = lane >> 4;   // which 16-lane half
    const int l     = lane & 15;
    const int mtile = wave & 1;    // 2 tiles cover R_DIM=32
    const int xtile = wave >> 1;   // 8 tiles cover X_DIM=128
    const int x0    = xtile * 16;
    const int m     = mtile * 16 + l;

    // LDS aperture occupies only the high 32 bits of a flat address, so the
    // low 32 bits of a flat shared pointer are the LDS byte offset.
    const unsigned xs_base = (unsigned)(unsigned long long)&xs[0][0][0];
    const unsigned rs_base = (unsigned)(unsigned long long)&rs[0][0][0];

    // Each thread stages 4 float4 of x and 1 float4 of r per chunk (5 async ops).
    // DMA destinations are padded-row addresses (per-lane LDS address VGPR).
    auto prefetch = [&](int buf, int kc) {
        const float* xsrc = xn + (kbase + kc) * XD;
#pragma unroll
        for (int i = 0; i < 4; ++i) {
            const int f   = tid + i * 512;        // float4 index within KC x XD
            const int row = f >> 5;               // 32 float4 per x row
            const int col = f & 31;
            async_b128(xs_base + (unsigned)buf * (KC * XP * 4)
                               + (unsigned)row * (XP * 4) + (unsigned)col * 16u,
                       xsrc + f * 4);
        }
        const float* rsrc = rn + (kbase + kc) * RD;
        {
            const int row = tid >> 3;             // 8 float4 per r row
            const int col = tid & 7;
            async_b128(rs_base + (unsigned)buf * (KC * RP * 4)
                               + (unsigned)row * (RP * 4) + (unsigned)col * 16u,
                       rsrc + tid * 4);
        }
    };

    prefetch(0, 0);
    int buf = 0;
    v8f acc = {};
    for (int kc = 0; kc < KPB; kc += KC) {
        const bool last = (kc + KC >= KPB);
        if (!last) {
            prefetch(buf ^ 1, kc + KC);  // overlap next chunk's DMA with compute
            asm volatile("s_wait_asynccnt 0x5" ::: "memory");  // current 5 done (in-order)
        } else {
            asm volatile("s_wait_asynccnt 0x0" ::: "memory");
        }
        __syncthreads();  // all waves' DMA for this buffer landed

#pragma unroll
        for (int kk = 0; kk < KC; kk += 4) {
            // A (16x4 f32): lanes 0-15 hold K={kk,kk+1}, lanes 16-31 hold K={kk+2,kk+3}
            v2f a, b;
            a.x = rs[buf][kk + 2 * half + 0][m];
            a.y = rs[buf][kk + 2 * half + 1][m];
            // B (4x16 f32): same K split across lane halves, N striped over lanes
            b.x = xs[buf][kk + 2 * half + 0][x0 + l];
            b.y = xs[buf][kk + 2 * half + 1][x0 + l];
            acc = __builtin_amdgcn_wmma_f32_16x16x4_f32(
                /*neg_a=*/false, a, /*neg_b=*/false, b,
                /*c_mod=*/(short)0, acc, /*reuse_a=*/false, /*reuse_b=*/false);
        }
        __syncthreads();  // everyone done reading this buffer before it refills
        buf ^= 1;
    }

    // C/D layout: VGPR i -> row (mtile*16 + half*8 + i), col x0+l
    float* yn = y + (long)n * RD * XD;
#pragma unroll
    for (int i = 0; i < 8; ++i) {
        const int row = mtile * 16 + half * 8 + i;
        atomicAdd(&yn[row * XD + (x0 + l)], acc[i]);
    }
}

// Steps 2+3: u = relu((y @ w1)/norm); out = einsum('ruvh,nrvh->nruv', w2, u)
__global__ __launch_bounds__(128) void ge_stage2(const float* __restrict__ y,
                                                 const float* __restrict__ w1,
                                                 const float* __restrict__ w2,
                                                 float* __restrict__ out) {
    __shared__ float yv[XD];
    __shared__ float us[VD * HD];  // u[v][h], index v*16+h

    const int nr = blockIdx.x;      // n*32 + r
    const int rr = nr & (RD - 1);
    const int t  = threadIdx.x;     // 128 threads

    yv[t] = y[(long)nr * XD + t];
    __syncthreads();

    // step 2: t = v*16 + h  (vh index), w1[x, r, v, h] -> x*4096 + r*128 + t
    const float* w1p = w1 + rr * (VD * HD) + t;
    float a = 0.0f;
#pragma unroll 8
    for (int xd = 0; xd < XD; ++xd) a = fmaf(yv[xd], w1p[xd * (RD * VD * HD)], a);
    a *= (1.0f / (float)KTOT);
    us[t] = fmaxf(a, 0.0f);
    __syncthreads();

    // step 3: t -> (u = t>>3, v = t&7); w2[r,u,v,h] -> r*2048 + u*128 + v*16 + h
    const int ui = t >> 3;
    const int v  = t & 7;
    const float* w2p = w2 + rr * (UD * VD * HD) + ui * (VD * HD) + v * HD;
    const float* up  = &us[v * HD];
    float o = 0.0f;
#pragma unroll
    for (int h = 0; h < HD; ++h) o = fmaf(w2p[h], up[h], o);
    out[(long)nr * (UD * VD) + ui * VD + v] = o;
}

extern "C" void kernel_launch(void* const* d_in, const int* in_sizes, int n_in,
                              void* d_out, int out_size, void* d_ws, size_t ws_size,
                              hipStream_t stream) {
    const float* x  = (const float*)d_in[0];
    const float* r  = (const float*)d_in[1];
    const float* w1 = (const float*)d_in[2];
    const float* w2 = (const float*)d_in[3];
    float* out = (float*)d_out;
    float* y   = (float*)d_ws;  // 32*32*128 floats = 512 KB partial-sum buffer

    const int yElems = NB * RD * XD;
    ge_zero<<<dim3((yElems + 255) / 256), 256, 0, stream>>>(y, yElems);
    ge_stage1<<<dim3(KSPLIT, NB), 512, 0, stream>>>(x, r, y);
    ge_stage2<<<dim3(NB * RD), 128, 0, stream>>>(y, w1, w2, out);
}